// HeteroLightGCN_59854664237672
// MI455X (gfx1250) — compile-verified
//
#include <hip/hip_runtime.h>

// HeteroLightGCN for MI455X (gfx1250): bandwidth/atomics-bound sparse message
// passing + WMMA (V_WMMA_F32_16X16X4_F32) for the dense D=64 -> 1 projection.

namespace {
constexpr int NUc = 200000;   // users
constexpr int NSc = 100000;   // spots
constexpr int Dc  = 64;       // embedding dim
constexpr int Ec  = 2000000;  // edges
}

// Both finalize launches rely on whole 16-row tiles (no per-row guards).
static_assert(NUc % 16 == 0 && NSc % 16 == 0, "tile assumption");

typedef __attribute__((ext_vector_type(2))) float v2f;
typedef __attribute__((ext_vector_type(8))) float v8f;

__device__ __forceinline__ void atomAddF(float* p, float v) {
  // Hardware GLOBAL_ATOMIC_ADD_F32 (non-returning) — no CAS loop.
  unsafeAtomicAdd(p, v);
}

// ---------------------------------------------------------------- zero fill
__global__ __launch_bounds__(256) void k_zero(float* __restrict__ p, long long n) {
  long long i4 = ((long long)blockIdx.x * blockDim.x + threadIdx.x) * 4;
  if (i4 >= n) return;
  if (i4 + 4 <= n) {
    float4 z; z.x = 0.f; z.y = 0.f; z.z = 0.f; z.w = 0.f;
    *reinterpret_cast<float4*>(p + i4) = z;
  } else {
    for (long long j = i4; j < n; ++j) p[j] = 0.f;
  }
}

// ---------------------------------------------------------------- degrees
__global__ __launch_bounds__(256) void k_degree(const int* __restrict__ esrc,
                                                const int* __restrict__ edst,
                                                float* __restrict__ deg_u,
                                                float* __restrict__ deg_s) {
  int e = blockIdx.x * blockDim.x + threadIdx.x;
  if (e >= Ec) return;
  atomAddF(&deg_u[esrc[e]], 1.0f);
  atomAddF(&deg_s[edst[e]], 1.0f);
}

// ---------------------------------------------------------------- inv_norm
__global__ __launch_bounds__(256) void k_invnorm(const int* __restrict__ esrc,
                                                 const int* __restrict__ edst,
                                                 const float* __restrict__ deg_u,
                                                 const float* __restrict__ deg_s,
                                                 float* __restrict__ invn) {
  int e = blockIdx.x * blockDim.x + threadIdx.x;
  if (e >= Ec) return;
  float du = deg_u[esrc[e]];
  float ds = deg_s[edst[e]];
  invn[e] = rsqrtf(du * ds);
}

// ------------------------------------------------- copy emb -> cur and acc
__global__ __launch_bounds__(256) void k_copy2(const float* __restrict__ src,
                                               float* __restrict__ dstA,
                                               float* __restrict__ dstB,
                                               long long n) {
  long long i4 = ((long long)blockIdx.x * blockDim.x + threadIdx.x) * 4;
  if (i4 >= n) return;
  if (i4 + 4 <= n) {
    float4 v = *reinterpret_cast<const float4*>(src + i4);
    *reinterpret_cast<float4*>(dstA + i4) = v;
    *reinterpret_cast<float4*>(dstB + i4) = v;
  } else {
    for (long long j = i4; j < n; ++j) { float v = src[j]; dstA[j] = v; dstB[j] = v; }
  }
}

// --------------------------------------- both-direction edge scatter (1 wave/edge)
__global__ __launch_bounds__(256) void k_scatter(const int* __restrict__ esrc,
                                                 const int* __restrict__ edst,
                                                 const float* __restrict__ invn,
                                                 const float* __restrict__ xu,
                                                 const float* __restrict__ xs,
                                                 float* __restrict__ new_u,
                                                 float* __restrict__ new_s) {
  long long tid = (long long)blockIdx.x * blockDim.x + threadIdx.x;
  long long e = tid >> 5;          // one wave32 per edge
  int lane = (int)(tid & 31);      // lane handles features [2*lane, 2*lane+1]
  if (e >= Ec) return;
  int s = esrc[e];
  int d = edst[e];
  float w = invn[e];
  size_t so = (size_t)s * Dc + 2 * lane;
  size_t dofs = (size_t)d * Dc + 2 * lane;
  // user -> spot
  float2 a = *reinterpret_cast<const float2*>(xu + so);
  atomAddF(new_s + dofs,     w * a.x);
  atomAddF(new_s + dofs + 1, w * a.y);
  // spot -> user (shared endpoints + normalization)
  float2 b = *reinterpret_cast<const float2*>(xs + dofs);
  atomAddF(new_u + so,     w * b.x);
  atomAddF(new_u + so + 1, w * b.y);
}

// ----------------------- cur = relu(new); acc += cur; new = 0  (fused sweep)
__global__ __launch_bounds__(256) void k_relu_acc_reset(float* __restrict__ neww,
                                                        float* __restrict__ cur,
                                                        float* __restrict__ acc,
                                                        long long n) {
  long long i4 = ((long long)blockIdx.x * blockDim.x + threadIdx.x) * 4;
  if (i4 >= n) return;
  if (i4 + 4 <= n) {
    float4 v = *reinterpret_cast<float4*>(neww + i4);
    float4 r;
    r.x = fmaxf(v.x, 0.f); r.y = fmaxf(v.y, 0.f);
    r.z = fmaxf(v.z, 0.f); r.w = fmaxf(v.w, 0.f);
    *reinterpret_cast<float4*>(cur + i4) = r;
    float4 s = *reinterpret_cast<float4*>(acc + i4);
    s.x += r.x; s.y += r.y; s.z += r.z; s.w += r.w;
    *reinterpret_cast<float4*>(acc + i4) = s;
    float4 z; z.x = 0.f; z.y = 0.f; z.z = 0.f; z.w = 0.f;
    *reinterpret_cast<float4*>(neww + i4) = z;
  } else {
    for (long long j = i4; j < n; ++j) {
      float r = fmaxf(neww[j], 0.f);
      cur[j] = r; acc[j] += r; neww[j] = 0.f;
    }
  }
}

// -------- finalize: x = acc/3 (written back), out = x @ W + b via WMMA f32
// One wave per 16-row tile (N is a multiple of 16 -> no row guards, EXEC all-1s).
// A (16x4 f32): lanes 0-15 = rows with K/K+1; lanes 16-31 carry K+2/K+3.
// B (4x16) = W replicated over all columns, so every column of C holds the
// same dot product. C layout: VGPR m -> row m (lanes 0-15) / row m+8
// (lanes 16-31); lanes 0 and 16 emit the 16 results.
// Phase structure (loads first, then WMMA chain) maximizes memory-level
// parallelism instead of one load-wait per WMMA.
__global__ __launch_bounds__(256) void k_finalize(float* __restrict__ acc,
                                                  const float* __restrict__ W,
                                                  const float* __restrict__ bias_p,
                                                  float* __restrict__ outp,
                                                  int N) {
  const int lane = threadIdx.x & 31;
  const long long wave = ((long long)blockIdx.x * blockDim.x + threadIdx.x) >> 5;
  const long long row0 = wave * 16;
  if (row0 >= N) return;                 // wave-uniform exit (full tiles only)
  const int myrow = (int)row0 + (lane & 15);
  const int koff = (lane < 16) ? 0 : 2;
  float* rp = acc + (size_t)myrow * Dc + koff;
  const float* wp = W + koff;
  const float scale = 1.0f / 3.0f;       // mean over [emb, layer1, layer2]

  // Phase 1: issue all 32 loads (16 row chunks + 16 W chunks) up front.
  float2 xa[16];
  float2 wb[16];
#pragma unroll
  for (int t = 0; t < 16; ++t) {
    xa[t] = *reinterpret_cast<const float2*>(rp + 4 * t);
    wb[t] = *reinterpret_cast<const float2*>(wp + 4 * t);
  }
  // Phase 2: normalize and write x back to d_out.
#pragma unroll
  for (int t = 0; t < 16; ++t) {
    xa[t].x *= scale;
    xa[t].y *= scale;
    *reinterpret_cast<float2*>(rp + 4 * t) = xa[t];
  }
  // Phase 3: 16 chained V_WMMA_F32_16X16X4_F32 over K = 64.
  v8f c = {};
#pragma unroll
  for (int t = 0; t < 16; ++t) {
    v2f av = {xa[t].x, xa[t].y};
    v2f bv = {wb[t].x, wb[t].y};
    c = __builtin_amdgcn_wmma_f32_16x16x4_f32(
        false, av, false, bv, (short)0, c, false, false);
  }
  const float bb = bias_p[0];
  if (lane == 0) {
#pragma unroll
    for (int m = 0; m < 8; ++m) outp[row0 + m] = c[m] + bb;
  } else if (lane == 16) {
#pragma unroll
    for (int m = 0; m < 8; ++m) outp[row0 + 8 + m] = c[m] + bb;
  }
}

// ---------------------------------------------------------------- launcher
extern "C" void kernel_launch(void* const* d_in, const int* in_sizes, int n_in,
                              void* d_out, int out_size, void* d_ws, size_t ws_size,
                              hipStream_t stream) {
  (void)in_sizes; (void)n_in; (void)out_size; (void)ws_size;
  const float* user_emb = (const float*)d_in[0];
  const float* spot_emb = (const float*)d_in[1];
  const float* W_user   = (const float*)d_in[2];
  const float* b_user   = (const float*)d_in[3];
  const float* W_spot   = (const float*)d_in[4];
  const float* b_spot   = (const float*)d_in[5];
  const int*   eu_src   = (const int*)d_in[6];
  const int*   eu_dst   = (const int*)d_in[7];

  float* out   = (float*)d_out;
  float* acc_u = out;                                   // x_user  [NU,64]
  float* acc_s = out + (size_t)NUc * Dc;                // x_spot  [NS,64]
  float* out_u = acc_s + (size_t)NSc * Dc;              // out_user[NU]
  float* out_s = out_u + NUc;                           // out_spot[NS]

  float* ws     = (float*)d_ws;
  float* deg_u  = ws;
  float* deg_s  = deg_u + NUc;
  float* xu_new = deg_s + NSc;
  float* xs_new = xu_new + (size_t)NUc * Dc;
  float* invn   = xs_new + (size_t)NSc * Dc;
  float* xu_cur = invn + Ec;
  float* xs_cur = xu_cur + (size_t)NUc * Dc;

  // zero [deg_u][deg_s][xu_new][xs_new] in one contiguous sweep
  const long long NZ = (long long)NUc + NSc + (long long)(NUc + NSc) * Dc;
  {
    long long t = (NZ + 3) / 4;
    k_zero<<<(unsigned)((t + 255) / 256), 256, 0, stream>>>(ws, NZ);
  }
  const unsigned eb = (Ec + 255) / 256;
  k_degree<<<eb, 256, 0, stream>>>(eu_src, eu_dst, deg_u, deg_s);
  k_invnorm<<<eb, 256, 0, stream>>>(eu_src, eu_dst, deg_u, deg_s, invn);

  k_copy2<<<(unsigned)(((long long)NUc * Dc / 4 + 255) / 256), 256, 0, stream>>>(
      user_emb, xu_cur, acc_u, (long long)NUc * Dc);
  k_copy2<<<(unsigned)(((long long)NSc * Dc / 4 + 255) / 256), 256, 0, stream>>>(
      spot_emb, xs_cur, acc_s, (long long)NSc * Dc);

  const long long scatThreads = (long long)Ec * 32;      // wave per edge
  const unsigned sb = (unsigned)((scatThreads + 255) / 256);
  const long long reluN = (long long)(NUc + NSc) * Dc;   // contiguous u|s sweep
  const unsigned rb = (unsigned)((reluN / 4 + 255) / 256);
  for (int layer = 0; layer < 2; ++layer) {
    k_scatter<<<sb, 256, 0, stream>>>(eu_src, eu_dst, invn, xu_cur, xs_cur,
                                      xu_new, xs_new);
    k_relu_acc_reset<<<rb, 256, 0, stream>>>(xu_new, xu_cur, acc_u, reluN);
  }

  k_finalize<<<(unsigned)(((long long)(NUc / 16) * 32 + 255) / 256), 256, 0, stream>>>(
      acc_u, W_user, b_user, out_u, NUc);
  k_finalize<<<(unsigned)(((long long)(NSc / 16) * 32 + 255) / 256), 256, 0, stream>>>(
      acc_s, W_spot, b_spot, out_s, NSc);
}